// GCNFeatureExtractor_37778532336358
// MI455X (gfx1250) — compile-verified
//
#include <hip/hip_runtime.h>
#include <hip/hip_bf16.h>

typedef float v2f __attribute__((ext_vector_type(2)));
typedef float v8f __attribute__((ext_vector_type(8)));

#define IN_DIM 256
#define HID    128
#define BN_EPS 1e-5f

// ---------------------------------------------------------------------------
// init: deg[i] = 1.0 (self-loop contribution), stats[0..255] = 0
// ---------------------------------------------------------------------------
__global__ __launch_bounds__(256) void init_misc(float* __restrict__ deg,
                                                 float* __restrict__ stats, int N) {
  int i = blockIdx.x * 256 + threadIdx.x;
  if (i < N)   deg[i]   = 1.0f;
  if (i < 256) stats[i] = 0.0f;
}

// ---------------------------------------------------------------------------
// GEMM h = x @ W using V_WMMA_F32_16X16X4_F32.
// Block: 256 threads = 8 waves. Block owns 16 rows of x (staged in LDS,
// stride 260 floats -> conflict-free, float4-aligned). Wave w computes the
// 16x16 output tile at columns [16w, 16w+16).
// ---------------------------------------------------------------------------
__global__ __launch_bounds__(256) void gemm_wmma(const float* __restrict__ x,
                                                 const float* __restrict__ W,
                                                 float* __restrict__ h) {
  __shared__ float sA[16 * 260];
  const int m0  = blockIdx.x * 16;
  const int tid = threadIdx.x;

  // cooperative load: 16 rows x 256 cols = 1024 float4s, coalesced
  #pragma unroll
  for (int i = tid; i < 16 * 64; i += 256) {
    const int r  = i >> 6;
    const int c4 = i & 63;
    const float4 v = reinterpret_cast<const float4*>(x + (size_t)(m0 + r) * IN_DIM)[c4];
    *reinterpret_cast<float4*>(&sA[r * 260 + c4 * 4]) = v;
  }
  __syncthreads();

  const int wave = tid >> 5;
  const int lane = tid & 31;
  const int half = lane >> 4;   // 0: K=0,1  1: K=2,3 (ISA 16x4 f32 A layout)
  const int l16  = lane & 15;
  const int n0   = wave * 16;

  const float* sArow = &sA[l16 * 260];
  v8f c = {};

  #pragma unroll 4
  for (int k0 = 0; k0 < IN_DIM; k0 += 4) {
    const int ka = k0 + 2 * half;
    v2f a; a.x = sArow[ka];                 a.y = sArow[ka + 1];
    v2f b; b.x = W[(size_t)ka * HID + n0 + l16];
           b.y = W[(size_t)(ka + 1) * HID + n0 + l16];
    c = __builtin_amdgcn_wmma_f32_16x16x4_f32(false, a, false, b,
                                              (short)0, c, false, false);
  }

  // C/D layout: vgpr r -> M = r + 8*half, N = l16
  float* out = h + (size_t)m0 * HID + n0 + l16;
  #pragma unroll
  for (int r = 0; r < 8; ++r)
    out[(size_t)(r + 8 * half) * HID] = c[r];
}

// ---------------------------------------------------------------------------
// degree count over edge targets
// ---------------------------------------------------------------------------
__global__ __launch_bounds__(256) void deg_count(const int* __restrict__ dst,
                                                 float* __restrict__ deg, int E) {
  int i = blockIdx.x * 256 + threadIdx.x;
  if (i < E) atomicAdd(&deg[dst[i]], 1.0f);
}

// in-place deg -> deg^{-1/2}
__global__ __launch_bounds__(256) void make_dinv(float* __restrict__ d, int N) {
  int i = blockIdx.x * 256 + threadIdx.x;
  if (i < N) {
    float v = d[i];
    d[i] = (v > 0.0f) ? rsqrtf(v) : 0.0f;
  }
}

// ---------------------------------------------------------------------------
// out[i,f] = h[i,f] * dinv[i]^2 + b[f]   (self-loop message + bias)
// ---------------------------------------------------------------------------
__global__ __launch_bounds__(256) void init_out(const float* __restrict__ h,
                                                const float* __restrict__ dinv,
                                                const float* __restrict__ b,
                                                float* __restrict__ out, int N) {
  size_t i = (size_t)blockIdx.x * 256 + threadIdx.x;
  if (i >= (size_t)N * HID) return;
  const int node = (int)(i >> 7);
  const int f    = (int)(i & (HID - 1));
  const float di = dinv[node];
  out[i] = h[i] * di * di + b[f];
}

// ---------------------------------------------------------------------------
// One wave32 per edge: lane L handles features [4L, 4L+4).
// Gather h[src] as float4 (512B coalesced), scale by dinv[s]*dinv[d],
// scatter-add into out[dst] with global_atomic_add_f32.
// ---------------------------------------------------------------------------
__global__ __launch_bounds__(256) void edge_agg(const int* __restrict__ src,
                                                const int* __restrict__ dst,
                                                const float* __restrict__ h,
                                                const float* __restrict__ dinv,
                                                float* __restrict__ out, int E) {
  const int wave = blockIdx.x * 8 + (threadIdx.x >> 5);
  if (wave >= E) return;
  const int lane = threadIdx.x & 31;
  const int s = src[wave];
  const int d = dst[wave];
  const float nrm = dinv[s] * dinv[d];
  const float4 v = reinterpret_cast<const float4*>(h + (size_t)s * HID)[lane];
  float* o = out + (size_t)d * HID + lane * 4;
  atomicAdd(o + 0, v.x * nrm);
  atomicAdd(o + 1, v.y * nrm);
  atomicAdd(o + 2, v.z * nrm);
  atomicAdd(o + 3, v.w * nrm);
}

// ---------------------------------------------------------------------------
// BN pass 1: per-feature sum and sum-of-squares (rows read coalesced: one
// 128-thread block reads whole 512B rows). Partial sums -> global atomics.
// ---------------------------------------------------------------------------
__global__ __launch_bounds__(128) void bn_stats(const float* __restrict__ out,
                                                float* __restrict__ stats, int N) {
  const int f = threadIdx.x;
  float s = 0.0f, s2 = 0.0f;
  for (int n = blockIdx.x; n < N; n += gridDim.x) {
    const float v = out[(size_t)n * HID + f];
    s  += v;
    s2 += v * v;
  }
  atomicAdd(&stats[f], s);
  atomicAdd(&stats[HID + f], s2);
}

// BN finalize: fold mean/var/gamma/beta into per-feature scale & shift
__global__ __launch_bounds__(128) void bn_finalize(const float* __restrict__ stats,
                                                   const float* __restrict__ gamma,
                                                   const float* __restrict__ beta,
                                                   float* __restrict__ ss, int N) {
  const int f = threadIdx.x;
  const float invN = 1.0f / (float)N;
  const float mean = stats[f] * invN;
  const float var  = stats[HID + f] * invN - mean * mean;
  const float istd = rsqrtf(var + BN_EPS);
  const float sc   = gamma[f] * istd;
  ss[f]       = sc;
  ss[HID + f] = beta[f] - mean * sc;
}

// BN pass 2 + ReLU, in place on d_out
__global__ __launch_bounds__(256) void bn_relu(float* __restrict__ out,
                                               const float* __restrict__ ss,
                                               size_t total) {
  size_t i = (size_t)blockIdx.x * 256 + threadIdx.x;
  if (i >= total) return;
  const int f = (int)(i & (HID - 1));
  const float v = out[i] * ss[f] + ss[HID + f];
  out[i] = fmaxf(v, 0.0f);
}

// ---------------------------------------------------------------------------
extern "C" void kernel_launch(void* const* d_in, const int* in_sizes, int n_in,
                              void* d_out, int out_size, void* d_ws, size_t ws_size,
                              hipStream_t stream) {
  const float* x     = (const float*)d_in[0];
  const int*   ei    = (const int*)d_in[1];   // [2, E] flat: src then dst
  const float* W     = (const float*)d_in[2];
  const float* b     = (const float*)d_in[3];
  const float* gamma = (const float*)d_in[4];
  const float* beta  = (const float*)d_in[5];
  float*       out   = (float*)d_out;

  const int N = in_sizes[0] / IN_DIM;   // 50000
  const int E = in_sizes[1] / 2;        // 600000
  const int* src = ei;
  const int* dst = ei + E;

  // workspace layout (floats)
  float* ws    = (float*)d_ws;
  float* h     = ws;                     // N*HID
  float* dinv  = h + (size_t)N * HID;    // N   (used first as deg)
  float* stats = dinv + N;               // 256 (sum | sumsq)
  float* ss    = stats + 256;            // 256 (scale | shift)

  const size_t total = (size_t)N * HID;

  // 1) deg=1 (self loops), stats=0
  init_misc<<<(N + 255) / 256, 256, 0, stream>>>(dinv, stats, N);
  // 2) h = x @ W  (WMMA f32)
  gemm_wmma<<<N / 16, 256, 0, stream>>>(x, W, h);
  // 3) degree over targets
  deg_count<<<(E + 255) / 256, 256, 0, stream>>>(dst, dinv, E);
  // 4) dinv = deg^{-1/2}
  make_dinv<<<(N + 255) / 256, 256, 0, stream>>>(dinv, N);
  // 5) out = h*dinv^2 + b (self-loop message)
  init_out<<<(int)((total + 255) / 256), 256, 0, stream>>>(h, dinv, b, out, N);
  // 6) edge scatter-add (1 wave per edge)
  edge_agg<<<(E + 7) / 8, 256, 0, stream>>>(src, dst, h, dinv, out, E);
  // 7) BN stats
  bn_stats<<<512, 128, 0, stream>>>(out, stats, N);
  // 8) fold into scale/shift
  bn_finalize<<<1, 128, 0, stream>>>(stats, gamma, beta, ss, N);
  // 9) normalize + ReLU in place
  bn_relu<<<(int)((total + 255) / 256), 256, 0, stream>>>(out, ss, total);
}